// GCN_39530878992718
// MI455X (gfx1250) — compile-verified
//
#include <hip/hip_runtime.h>
#include <hip/hip_bf16.h>

typedef __attribute__((ext_vector_type(2))) float v2f;
typedef __attribute__((ext_vector_type(8))) float v8f;

// ---------------------------------------------------------------- degree prep
__global__ void k_fill_deg(float* __restrict__ deg, int N) {
    int i = blockIdx.x * blockDim.x + threadIdx.x;
    if (i < N) deg[i] = 1.0f;   // self-loop contribution
}

__global__ void k_deg_accum(const int* __restrict__ dst, float* __restrict__ deg, int E) {
    int e = blockIdx.x * blockDim.x + threadIdx.x;
    if (e < E) unsafeAtomicAdd(&deg[dst[e]], 1.0f);
}

__global__ void k_rsqrt(float* __restrict__ deg, int N) {
    int i = blockIdx.x * blockDim.x + threadIdx.x;
    if (i < N) {
        float d = deg[i];
        deg[i] = d > 0.0f ? rsqrtf(d) : 0.0f;
    }
}

// ---------------------------------------------------------------- WMMA GEMM
// out[N x dout] = act(A)[N x din] @ W[din x dout], act = ReLU if RELU.
// One wave computes one 16x16 output tile using V_WMMA_F32_16X16X4_F32.
// Inner loop is branchless: EXEC stays all-ones across every WMMA (ISA req),
// B loads use a strided pointer bump (no per-iter 64-bit muls), and the
// out-of-range column tile (dout=40 case) is handled by clamp + x0.0 mask
// (exact: masked lanes are never stored).
template <bool RELU>
__global__ void k_gemm_wmma(const float* __restrict__ A, const float* __restrict__ W,
                            float* __restrict__ out, int N, int din, int dout) {
    int wave = (int)((blockIdx.x * blockDim.x + threadIdx.x) >> 5);
    int lane = threadIdx.x & 31;
    int tiles_n = (dout + 15) >> 4;
    int tile_m  = wave / tiles_n;
    int tile_n  = wave - tile_m * tiles_n;
    if (tile_m * 16 >= N) return;           // uniform per-wave guard

    int m0 = tile_m << 4, n0 = tile_n << 4;
    int lane16 = lane & 15;
    int hi     = lane >> 4;                 // 0: K pair {0,1}, 1: K pair {2,3}
    int row    = m0 + lane16;               // A-matrix M for this lane
    if (row >= N) row = N - 1;              // safe clamp (N is multiple of 16 here)
    int ncol   = n0 + lane16;               // B/D N for this lane
    bool colOK = ncol < dout;
    int ncolC  = colOK ? ncol : (dout - 1); // clamped (in-range) column
    float bmask = colOK ? 1.0f : 0.0f;      // zero out padded columns

    v8f acc = {0.f, 0.f, 0.f, 0.f, 0.f, 0.f, 0.f, 0.f};
    const float* arow = A + (size_t)row * din + (hi << 1);
    const float* wp   = W + (size_t)(hi << 1) * dout + ncolC;
    const int wstep = dout * 4;

    for (int k = 0; k < din; k += 4) {
        float a0 = arow[k];
        float a1 = arow[k + 1];
        if (RELU) { a0 = fmaxf(a0, 0.f); a1 = fmaxf(a1, 0.f); }
        v2f a; a.x = a0; a.y = a1;

        v2f b;
        b.x = wp[0]    * bmask;
        b.y = wp[dout] * bmask;
        wp += wstep;

        // D = A x B + C, 16x16x4 f32 (exact f32 math)
        acc = __builtin_amdgcn_wmma_f32_16x16x4_f32(
            /*neg_a=*/false, a, /*neg_b=*/false, b,
            /*c_mod=*/(short)0, acc, /*reuse_a=*/false, /*reuse_b=*/false);
    }

    if (colOK) {
        int mbase = m0 + (hi << 3);         // VGPR r holds row mbase + r
#pragma unroll
        for (int r = 0; r < 8; ++r) {
            int m = mbase + r;
            if (m < N) out[(size_t)m * dout + ncol] = acc[r];
        }
    }
}

// ------------------------------------------------- self-loop init of aggregate
// D is compile-time so i/D is a mul-shift, not an integer-division sequence.
template <int D>
__global__ void k_init_agg(const float* __restrict__ hlin, const float* __restrict__ dinv,
                           float* __restrict__ agg, int total) {
    int i = blockIdx.x * blockDim.x + threadIdx.x;
    if (i < total) {
        float w = dinv[i / D];
        agg[i] = hlin[i] * w * w;           // self-loop: norm = dinv[i]^2
    }
}

// ------------------------------------------------- edge gather + atomic scatter
// thread = (edge, 4-float chunk); chunk-minor so an edge's D/4 threads read
// D contiguous floats (L2-resident gather: the 25.6MB feature matrix sits in
// the 192MB L2), then 4 hardware global_atomic_add_f32 ops.
template <int D>
__global__ void k_scatter(const int* __restrict__ src, const int* __restrict__ dst,
                          const float* __restrict__ dinv, const float* __restrict__ hlin,
                          float* __restrict__ agg, int E) {
    constexpr int CH = D >> 2;
    int idx = blockIdx.x * blockDim.x + threadIdx.x;
    int e = idx / CH;                       // constant divisor -> mul/shift
    if (e >= E) return;
    int c = idx - e * CH;

    int s = src[e], t = dst[e];
    float ns = dinv[s] * dinv[t];

    const float4* hv = (const float4*)(hlin + (size_t)s * D);
    float4 v = hv[c];
    float* o = agg + (size_t)t * D + (c << 2);
    unsafeAtomicAdd(o + 0, v.x * ns);
    unsafeAtomicAdd(o + 1, v.y * ns);
    unsafeAtomicAdd(o + 2, v.z * ns);
    unsafeAtomicAdd(o + 3, v.w * ns);
}

__global__ void k_relu(float* __restrict__ p, int n) {
    int i = blockIdx.x * blockDim.x + threadIdx.x;
    if (i < n) p[i] = fmaxf(p[i], 0.f);
}

// ---------------------------------------------------------------- launcher
extern "C" void kernel_launch(void* const* d_in, const int* in_sizes, int n_in,
                              void* d_out, int out_size, void* d_ws, size_t ws_size,
                              hipStream_t stream) {
    const float* x  = (const float*)d_in[0];
    const int*   ei = (const int*)d_in[1];    // edge_index, [2, E] flattened
    const float* W0 = (const float*)d_in[2];  // [128, 64]
    const float* W1 = (const float*)d_in[3];  // [64, 64]
    const float* W2 = (const float*)d_in[4];  // [64, 40]
    float* out = (float*)d_out;

    const int FIN = 128, H = 64, C = 40;
    const int E = in_sizes[1] / 2;
    const int N = in_sizes[0] / FIN;
    const int* srcp = ei;
    const int* dstp = ei + E;

    float* ws   = (float*)d_ws;
    float* dinv = ws;                         // N floats (deg, then rsqrt in place)
    float* hlin = ws + (1 << 17);             // N*64 floats (linear output)
    float* hagg = hlin + (size_t)N * H;       // N*64 floats (aggregate)

    auto cdiv = [](long long a, long long b) { return (int)((a + b - 1) / b); };
    const int T = 256;

    // --- normalization: deg = 1 + in-degree, dinv = rsqrt(deg) ---
    k_fill_deg <<<cdiv(N, T), T, 0, stream>>>(dinv, N);
    k_deg_accum<<<cdiv(E, T), T, 0, stream>>>(dstp, dinv, E);
    k_rsqrt    <<<cdiv(N, T), T, 0, stream>>>(dinv, N);

    long long tiles_m = (N + 15) / 16;

    // --- layer 0: x @ W0, aggregate ---
    {
        long long waves = tiles_m * ((H + 15) / 16);
        k_gemm_wmma<false><<<cdiv(waves * 32, T), T, 0, stream>>>(x, W0, hlin, N, FIN, H);
        k_init_agg<64><<<cdiv((long long)N * H, T), T, 0, stream>>>(hlin, dinv, hagg, N * H);
        k_scatter<64><<<cdiv((long long)E * (H / 4), T), T, 0, stream>>>(srcp, dstp, dinv, hlin, hagg, E);
    }
    // --- layer 1: relu of layer-0 aggregate fused into A-load ---
    {
        long long waves = tiles_m * ((H + 15) / 16);
        k_gemm_wmma<true><<<cdiv(waves * 32, T), T, 0, stream>>>(hagg, W1, hlin, N, H, H);
        k_init_agg<64><<<cdiv((long long)N * H, T), T, 0, stream>>>(hlin, dinv, hagg, N * H);
        k_scatter<64><<<cdiv((long long)E * (H / 4), T), T, 0, stream>>>(srcp, dstp, dinv, hlin, hagg, E);
    }
    // --- layer 2: aggregate straight into d_out, relu in place ---
    {
        long long waves = tiles_m * ((C + 15) / 16);
        k_gemm_wmma<true><<<cdiv(waves * 32, T), T, 0, stream>>>(hagg, W2, hlin, N, H, C);
        k_init_agg<40><<<cdiv((long long)N * C, T), T, 0, stream>>>(hlin, dinv, out, N * C);
        k_scatter<40><<<cdiv((long long)E * (C / 4), T), T, 0, stream>>>(srcp, dstp, dinv, hlin, out, E);
        k_relu    <<<cdiv((long long)N * C, T), T, 0, stream>>>(out, N * C);
    }
}